// FNO_block_54589034332871
// MI455X (gfx1250) — compile-verified
//
#include <hip/hip_runtime.h>
#include <hip/hip_bf16.h>

// ---------------------------------------------------------------------------
// FNO block on gfx1250 (MI455X). All contractions (truncated DFTs, spectral
// channel mix, pointwise MLP) run on v_wmma_f32_16x16x4_f32. The W-DFT twiddle
// matrix is staged to LDS with the Tensor Data Mover (tensor_load_to_lds +
// s_wait_tensorcnt). GroupNorm statistics are fused into producer epilogues.
// d_out doubles as the 134MB spatial scratch (3 in-place passes).
// ---------------------------------------------------------------------------

#define Bn 8
#define Cn 64
#define Hn 256
#define Wn 256
#define HWn 65536
#define M2n 32

typedef float v2f __attribute__((ext_vector_type(2)));
typedef float v8f __attribute__((ext_vector_type(8)));
typedef unsigned int v4u __attribute__((ext_vector_type(4)));
typedef int v4i __attribute__((ext_vector_type(4)));
typedef int v8i __attribute__((ext_vector_type(8)));

__device__ __forceinline__ v8f wmma4(v2f a, v2f b, v8f c) {
  return __builtin_amdgcn_wmma_f32_16x16x4_f32(false, a, false, b, (short)0, c,
                                               false, false);
}

__device__ __forceinline__ float gelu_exact(float v) {
  return 0.5f * v * (1.0f + erff(v * 0.70710678118654752440f));
}

// 1-D TDM copy: nel 4-byte elements, global -> LDS (6-arg clang-23 builtin).
__device__ __forceinline__ void tdm_load_1d(void* lds, const void* gsrc,
                                            unsigned nel) {
  unsigned lo = (unsigned)(size_t)lds;  // low 32 bits of generic ptr = LDS off
  unsigned long long ga = (unsigned long long)(size_t)gsrc;
  v4u g0;
  g0.x = 1u;                                        // count=1, user descriptor
  g0.y = lo;                                        // lds_addr
  g0.z = (unsigned)ga;                              // global_addr[31:0]
  g0.w = (unsigned)((ga >> 32) & 0x1FFFFFFull) | (2u << 30);  // [56:32]|type=2
  v8i g1;
  g1[0] = (int)(2u << 16);                          // data_size = 4B
  g1[1] = (int)((nel & 0xFFFFu) << 16);             // tensor_dim0[15:0]
  g1[2] = (int)(((nel >> 16) & 0xFFFFu) | (1u << 16));  // dim0 hi | dim1=1
  g1[3] = (int)((nel & 0xFFFFu) << 16);             // tile_dim0
  g1[4] = 0;                                        // tile_dim1/2 unused
  g1[5] = (int)nel;                                 // tensor_dim0_stride lo
  g1[6] = 0;
  g1[7] = 0;
  v4i z4 = {0, 0, 0, 0};
  v8i z8 = {0, 0, 0, 0, 0, 0, 0, 0};
  __builtin_amdgcn_tensor_load_to_lds(g0, g1, z4, z4, z8, 0);
}

// ---------------------------------------------------------------------------
// Kernel 0: twiddle/tables + zero the fused-GN statistic accumulators.
// TW  [256][64]: forward W-DFT  (cols 0..31 re = cos, 32..63 im = -sin)
// T2  [64][512]: forward H-DFT over complex rows (even k: cos, odd k: sin)
// TIH [256][128]: inverse H-DFT (cos/H, sin/H)
// TIW [64][256]: inverse real W-DFT (Hermitian fold: 2x for m>0)
// ---------------------------------------------------------------------------
__global__ __launch_bounds__(256) void k_tables(float* __restrict__ ws) {
  float* TW = ws;
  float* T2 = ws + 16384;
  float* TIH = ws + 49152;
  float* TIW = ws + 81920;
  float* stats = ws + 98304 + 8388608 + 2097152 + 2097152;
  const float step = 6.28318530717958647692f / 256.0f;
  int tid = threadIdx.x;
  {  // TW: one row per thread
    int w = tid;
    for (int m = 0; m < 32; ++m) {
      float th = (float)((w * m) & 255) * step;
      TW[w * 64 + m] = __cosf(th);
      TW[w * 64 + 32 + m] = -__sinf(th);
    }
  }
  for (int e = tid; e < 64 * 256; e += 256) {  // T2
    int r = e >> 8, h = e & 255;
    int kp = (r < 32) ? r : r + 192;  // kept H modes: 0..31 and 224..255
    float th = (float)((h * kp) & 255) * step;
    T2[r * 512 + 2 * h] = __cosf(th);
    T2[r * 512 + 2 * h + 1] = __sinf(th);
  }
  for (int e = tid; e < 256 * 64; e += 256) {  // TIH
    int h = e >> 6, k = e & 63;
    int kp = (k < 32) ? k : k + 192;
    float th = (float)((h * kp) & 255) * step;
    TIH[h * 128 + 2 * k] = __cosf(th) * (1.0f / 256.0f);
    TIH[h * 128 + 2 * k + 1] = __sinf(th) * (1.0f / 256.0f);
  }
  for (int e = tid; e < 64 * 256; e += 256) {  // TIW
    int k = e >> 8, w = e & 255;
    int m = k & 31;
    float sc = ((m == 0) ? 1.0f : 2.0f) * (1.0f / 256.0f);
    float th = (float)((w * m) & 255) * step;
    TIW[k * 256 + w] = (k < 32) ? sc * __cosf(th) : -sc * __sinf(th);
  }
  if (tid < 32) stats[tid] = 0.0f;
}

// ---------------------------------------------------------------------------
// GEMM1: forward DFT along W.  Xw[row=(b,c,h)][64] = x[row][0..255] * TW.
// TW staged into LDS by the Tensor Data Mover (64KB; WGP has 320KB LDS).
// Grid 2048 x 128 threads; each wave: 16 rows x 64 cols, K=256.
// ---------------------------------------------------------------------------
__global__ __launch_bounds__(128) void k_dftw(const float* __restrict__ x,
                                              const float* __restrict__ TW,
                                              float* __restrict__ Xw) {
  __shared__ float TWs[16384];
  int tid = threadIdx.x;
  if (tid < 32) {  // wave 0 issues the TDM copy, waits TENSORcnt
    tdm_load_1d(TWs, TW, 16384u);
    __builtin_amdgcn_s_wait_tensorcnt(0);
  }
  __syncthreads();
  int wave = tid >> 5, lane = tid & 31;
  int row0 = blockIdx.x * 64 + wave * 16;
  const float* A = x + (size_t)row0 * 256;
  int r = lane & 15, koff = (lane >> 4) << 1, colb = lane & 15;
  v8f acc[4] = {};
  for (int k0 = 0; k0 < 256; k0 += 4) {
    int kk = k0 + koff;
    v2f a = {A[r * 256 + kk], A[r * 256 + kk + 1]};
#pragma unroll
    for (int t = 0; t < 4; ++t) {
      const float* Bp = TWs + t * 16 + colb;
      v2f b = {Bp[kk * 64], Bp[(kk + 1) * 64]};
      acc[t] = wmma4(a, b, acc[t]);
    }
  }
  float* O = Xw + (size_t)row0 * 64;
  int rb = (lane >> 4) * 8;
#pragma unroll
  for (int t = 0; t < 4; ++t)
#pragma unroll
    for (int v = 0; v < 8; ++v) O[(rb + v) * 64 + t * 16 + colb] = acc[t][v];
}

// ---------------------------------------------------------------------------
// GEMM2: forward DFT along H (complex). Per (b,c): Xf[64 modes][64] =
// T2(64x512) * Z(512x64), where Z's odd rows (-i*x) are synthesized on the fly
// from Xw (re/im half swap with negation). Grid 512 x 128.
// ---------------------------------------------------------------------------
__global__ __launch_bounds__(128) void k_dfth(const float* __restrict__ Xw,
                                              const float* __restrict__ T2,
                                              float* __restrict__ Xf) {
  int bc = blockIdx.x;
  int tid = threadIdx.x, wave = tid >> 5, lane = tid & 31;
  int r = lane & 15, koff = (lane >> 4) << 1, colb = lane & 15;
  const float* Xb = Xw + (size_t)bc * 256 * 64;
  const float* A = T2 + (size_t)(wave * 16) * 512;
  v8f acc[4] = {};
  for (int k0 = 0; k0 < 512; k0 += 4) {
    int kk = k0 + koff;
    v2f a = {A[r * 512 + kk], A[r * 512 + kk + 1]};
    const float* Xrow = Xb + (kk >> 1) * 64;  // kk even: pair (2h, 2h+1)
#pragma unroll
    for (int t = 0; t < 4; ++t) {
      int col = t * 16 + colb;
      float bx = Xrow[col];
      float by = (col < 32) ? Xrow[col + 32] : -Xrow[col - 32];  // -i * x
      v2f b = {bx, by};
      acc[t] = wmma4(a, b, acc[t]);
    }
  }
  float* O = Xf + (size_t)bc * 4096 + (size_t)(wave * 16) * 64;
  int rb = (lane >> 4) * 8;
#pragma unroll
  for (int t = 0; t < 4; ++t)
#pragma unroll
    for (int v = 0; v < 8; ++v) O[(rb + v) * 64 + t * 16 + colb] = acc[t][v];
}

// ---------------------------------------------------------------------------
// Spectral channel mix: per mode (2048 blocks), complex 64x64 weight applied
// to all 8 batches. Real GEMM M=64, N=16 (b x re/im), K=128 (2*C). Weight
// (re,im) interleave read directly from w{1,2}{r,i}; B staged to LDS with the
// complex expansion. 4 waves, one 16x16 tile each.
// ---------------------------------------------------------------------------
__global__ __launch_bounds__(128) void k_spec(const float* __restrict__ Xf,
                                              const float* __restrict__ w1r,
                                              const float* __restrict__ w1i,
                                              const float* __restrict__ w2r,
                                              const float* __restrict__ w2i,
                                              float* __restrict__ Yc) {
  __shared__ float Bs[128 * 16];
  int mid = blockIdx.x, mh = mid >> 5, m2 = mid & 31;
  int tid = threadIdx.x;
  {  // build B: rows 2i = (Xr | Xi), rows 2i+1 = (-Xi | Xr)
    int k = tid, i = k >> 1, odd = k & 1;
    for (int n = 0; n < 8; ++n) {
      const float* Xr = Xf + (((size_t)(n * 64 + i) * 64 + mh) << 6);
      float re = Xr[m2], im = Xr[32 + m2];
      if (!odd) {
        Bs[k * 16 + n] = re;
        Bs[k * 16 + 8 + n] = im;
      } else {
        Bs[k * 16 + n] = -im;
        Bs[k * 16 + 8 + n] = re;
      }
    }
  }
  __syncthreads();
  const float* wr = (mh < 32) ? w1r : w2r;
  const float* wi = (mh < 32) ? w1i : w2i;
  int xx = mh & 31;
  int wave = tid >> 5, lane = tid & 31;
  int o = wave * 16 + (lane & 15);
  int koff = (lane >> 4) << 1, colb = lane & 15;
  v8f acc = {};
  for (int k0 = 0; k0 < 128; k0 += 4) {
    int kk = k0 + koff, i = kk >> 1;
    size_t widx = ((((size_t)i * 64 + o) * 32 + xx) << 5) + m2;
    v2f a = {wr[widx], wi[widx]};  // A[o][2i]=Wr, A[o][2i+1]=Wi
    v2f b = {Bs[kk * 16 + colb], Bs[(kk + 1) * 16 + colb]};
    acc = wmma4(a, b, acc);
  }
  float* O = Yc + (size_t)mid * 1024 + (size_t)(wave * 16) * 16;
  int rb = (lane >> 4) * 8;
#pragma unroll
  for (int v = 0; v < 8; ++v) O[(rb + v) * 16 + colb] = acc[v];
}

// ---------------------------------------------------------------------------
// GEMM3: inverse DFT along H. Per (b,o): Yh[256][64] = TIH(256x128) * Zc,
// Zc (odd rows = +i*x) synthesized from Yc into LDS (32KB). Grid 2048 x 128.
// ---------------------------------------------------------------------------
__global__ __launch_bounds__(128) void k_idfth(const float* __restrict__ Yc,
                                               const float* __restrict__ TIH,
                                               float* __restrict__ Yh) {
  __shared__ float Bs[128 * 64];
  int bc = blockIdx.x >> 2, hblk = blockIdx.x & 3;
  int b = bc >> 6, o = bc & 63;
  int tid = threadIdx.x;
  {
    int k = tid, q = k >> 1, odd = k & 1;
    for (int m2 = 0; m2 < 32; ++m2) {
      const float* Yp = Yc + (size_t)(q * 32 + m2) * 1024 + o * 16 + b;
      float re = Yp[0], im = Yp[8];
      if (!odd) {
        Bs[k * 64 + m2] = re;
        Bs[k * 64 + 32 + m2] = im;
      } else {  // +i * x : re col = -im, im col = +re
        Bs[k * 64 + m2] = -im;
        Bs[k * 64 + 32 + m2] = re;
      }
    }
  }
  __syncthreads();
  int wave = tid >> 5, lane = tid & 31;
  int h0 = hblk * 64 + wave * 16;
  const float* A = TIH + (size_t)h0 * 128;
  int r = lane & 15, koff = (lane >> 4) << 1, colb = lane & 15;
  v8f acc[4] = {};
  for (int k0 = 0; k0 < 128; k0 += 4) {
    int kk = k0 + koff;
    v2f a = {A[r * 128 + kk], A[r * 128 + kk + 1]};
#pragma unroll
    for (int t = 0; t < 4; ++t) {
      v2f bfr = {Bs[kk * 64 + t * 16 + colb], Bs[(kk + 1) * 64 + t * 16 + colb]};
      acc[t] = wmma4(a, bfr, acc[t]);
    }
  }
  float* O = Yh + ((size_t)bc * 256 + h0) * 64;
  int rb = (lane >> 4) * 8;
#pragma unroll
  for (int t = 0; t < 4; ++t)
#pragma unroll
    for (int v = 0; v < 8; ++v) O[(rb + v) * 64 + t * 16 + colb] = acc[t][v];
}

// ---------------------------------------------------------------------------
// GEMM4: inverse real DFT along W + fused GroupNorm-1 statistics.
// y[row][0..255] = Yh[row][0..63] * TIW. Writes directly into d_out.
// Grid 8192 x 128 (16 rows/block); per-block reduce -> atomicAdd(stats[b]).
// ---------------------------------------------------------------------------
__global__ __launch_bounds__(128) void k_idftw(const float* __restrict__ Yh,
                                               const float* __restrict__ TIW,
                                               float* __restrict__ y,
                                               float* __restrict__ stats) {
  int tid = threadIdx.x, wave = tid >> 5, lane = tid & 31;
  size_t row0 = (size_t)blockIdx.x * 16;
  int b = blockIdx.x >> 10;
  const float* A = Yh + row0 * 64;
  int r = lane & 15, koff = (lane >> 4) << 1, colb = lane & 15;
  v8f acc[4] = {};
  for (int k0 = 0; k0 < 64; k0 += 4) {
    int kk = k0 + koff;
    v2f a = {A[r * 64 + kk], A[r * 64 + kk + 1]};
#pragma unroll
    for (int t = 0; t < 4; ++t) {
      int c0 = wave * 64 + t * 16 + colb;
      v2f bfr = {TIW[kk * 256 + c0], TIW[(kk + 1) * 256 + c0]};
      acc[t] = wmma4(a, bfr, acc[t]);
    }
  }
  float s = 0.0f, q = 0.0f;
  int rb = (lane >> 4) * 8;
#pragma unroll
  for (int t = 0; t < 4; ++t)
#pragma unroll
    for (int v = 0; v < 8; ++v) {
      float val = acc[t][v];
      y[(row0 + rb + v) * 256 + wave * 64 + t * 16 + colb] = val;
      s += val;
      q += val * val;
    }
  __shared__ float rs[128], rq[128];
  rs[tid] = s;
  rq[tid] = q;
  __syncthreads();
  for (int st = 64; st > 0; st >>= 1) {
    if (tid < st) {
      rs[tid] += rs[tid + st];
      rq[tid] += rq[tid + st];
    }
    __syncthreads();
  }
  if (tid == 0) {
    atomicAdd(&stats[b], rs[0]);
    atomicAdd(&stats[8 + b], rq[0]);
  }
}

// ---------------------------------------------------------------------------
// FiLM gamma/beta from t (tiny).
// ---------------------------------------------------------------------------
__global__ __launch_bounds__(512) void k_film(const float* __restrict__ t,
                                              const float* __restrict__ gw,
                                              const float* __restrict__ gb,
                                              const float* __restrict__ bw,
                                              const float* __restrict__ bb,
                                              float* __restrict__ gbuf) {
  int tid = threadIdx.x;
  int b = tid >> 6, o = tid & 63;
  float g = gb[o], be = bb[o];
  for (int c = 0; c < 64; ++c) {
    float tv = t[b * 64 + c];
    g += gw[o * 64 + c] * tv;
    be += bw[o * 64 + c] * tv;
  }
  gbuf[tid] = g;
  gbuf[512 + tid] = be;
}

// ---------------------------------------------------------------------------
// Fused pointwise stage, in place on d_out:
//   u = gelu(GN1(y)*w + b + x);  h = gelu(W1 u + b1);  y2 = W2 h + b2
// Channel GEMMs via WMMA from LDS tiles (64 channels x 64 pixels).
// Also accumulates GroupNorm-2 statistics. Grid 8192 x 128.
// ---------------------------------------------------------------------------
__global__ __launch_bounds__(128) void k_mlp(
    const float* __restrict__ xin, const float* __restrict__ n1w,
    const float* __restrict__ n1b, const float* __restrict__ w1,
    const float* __restrict__ b1, const float* __restrict__ w2,
    const float* __restrict__ b2, float* __restrict__ stats,
    float* __restrict__ y) {
  __shared__ float U[64 * 64];
  __shared__ float Hm[32 * 64];
  __shared__ float W1s[32 * 64];
  __shared__ float W2s[64 * 32];
  __shared__ float rs[128], rq[128];
  int tid = threadIdx.x;
  int b = blockIdx.x >> 10;
  size_t base = (size_t)b * 4194304 + ((size_t)(blockIdx.x & 1023)) * 64;
  float mu1 = stats[b] * (1.0f / 4194304.0f);
  float var1 = stats[8 + b] * (1.0f / 4194304.0f) - mu1 * mu1;
  float rstd1 = rsqrtf(var1 + 1e-5f);
  for (int i = tid; i < 2048; i += 128) W1s[i] = w1[i];
  for (int i = tid; i < 2048; i += 128) W2s[i] = w2[i];
  for (int idx = tid; idx < 4096; idx += 128) {
    int c = idx >> 6, p = idx & 63;
    size_t g = base + (size_t)c * 65536 + p;
    float u = (y[g] - mu1) * rstd1 * n1w[c] + n1b[c];
    U[c * 64 + p] = gelu_exact(u + xin[g]);
  }
  __syncthreads();
  int wave = tid >> 5, lane = tid & 31;
  int r = lane & 15, koff = (lane >> 4) << 1, colb = lane & 15;
  int rb = (lane >> 4) * 8;
  {  // MLP1: 32x64 = W1(32x64) @ U(64x64), gelu, to LDS
    int mt = wave & 1;
    int nt0 = (wave >> 1) * 2;
    v8f acc[2] = {};
    for (int k0 = 0; k0 < 64; k0 += 4) {
      int kk = k0 + koff;
      v2f a = {W1s[(mt * 16 + r) * 64 + kk], W1s[(mt * 16 + r) * 64 + kk + 1]};
#pragma unroll
      for (int j = 0; j < 2; ++j) {
        int c0 = (nt0 + j) * 16 + colb;
        v2f bf = {U[kk * 64 + c0], U[(kk + 1) * 64 + c0]};
        acc[j] = wmma4(a, bf, acc[j]);
      }
    }
#pragma unroll
    for (int j = 0; j < 2; ++j)
#pragma unroll
      for (int v = 0; v < 8; ++v) {
        int row = mt * 16 + rb + v;
        Hm[row * 64 + (nt0 + j) * 16 + colb] = gelu_exact(acc[j][v] + b1[row]);
      }
  }
  __syncthreads();
  float s = 0.0f, q = 0.0f;
  {  // MLP2: 64x64 = W2(64x32) @ Hm, + bias, write back + GN2 stats
    v8f acc[4] = {};
    for (int k0 = 0; k0 < 32; k0 += 4) {
      int kk = k0 + koff;
      v2f a = {W2s[(wave * 16 + r) * 32 + kk],
               W2s[(wave * 16 + r) * 32 + kk + 1]};
#pragma unroll
      for (int t = 0; t < 4; ++t) {
        int c0 = t * 16 + colb;
        v2f bf = {Hm[kk * 64 + c0], Hm[(kk + 1) * 64 + c0]};
        acc[t] = wmma4(a, bf, acc[t]);
      }
    }
#pragma unroll
    for (int t = 0; t < 4; ++t)
#pragma unroll
      for (int v = 0; v < 8; ++v) {
        int row = wave * 16 + rb + v;
        float val = acc[t][v] + b2[row];
        y[base + (size_t)row * 65536 + t * 16 + colb] = val;
        s += val;
        q += val * val;
      }
  }
  rs[tid] = s;
  rq[tid] = q;
  __syncthreads();
  for (int st = 64; st > 0; st >>= 1) {
    if (tid < st) {
      rs[tid] += rs[tid + st];
      rq[tid] += rq[tid + st];
    }
    __syncthreads();
  }
  if (tid == 0) {
    atomicAdd(&stats[16 + b], rs[0]);
    atomicAdd(&stats[24 + b], rq[0]);
  }
}

// ---------------------------------------------------------------------------
// Finalize in place: out = gamma*GN2(y2) + beta + gelu(x). float4 streams.
// ---------------------------------------------------------------------------
__global__ __launch_bounds__(256) void k_final(const float* __restrict__ xin,
                                               const float* __restrict__ stats,
                                               const float* __restrict__ gbuf,
                                               float* __restrict__ out) {
  size_t i4 = (size_t)blockIdx.x * 256 + threadIdx.x;
  size_t base = i4 * 4;
  int plane = (int)(base >> 16);
  int b = plane >> 6, c = plane & 63;
  float mu = stats[16 + b] * (1.0f / 4194304.0f);
  float var = stats[24 + b] * (1.0f / 4194304.0f) - mu * mu;
  float rstd = rsqrtf(var + 1e-5f);
  float g = gbuf[b * 64 + c], be = gbuf[512 + b * 64 + c];
  float4 v = *(const float4*)(out + base);
  float4 xv = *(const float4*)(xin + base);
  float4 o;
  o.x = g * ((v.x - mu) * rstd) + be + gelu_exact(xv.x);
  o.y = g * ((v.y - mu) * rstd) + be + gelu_exact(xv.y);
  o.z = g * ((v.z - mu) * rstd) + be + gelu_exact(xv.z);
  o.w = g * ((v.w - mu) * rstd) + be + gelu_exact(xv.w);
  *(float4*)(out + base) = o;
}

extern "C" void kernel_launch(void* const* d_in, const int* in_sizes, int n_in,
                              void* d_out, int out_size, void* d_ws,
                              size_t ws_size, hipStream_t stream) {
  (void)in_sizes;
  (void)n_in;
  (void)out_size;
  (void)ws_size;
  const float* x = (const float*)d_in[0];
  const float* t = (const float*)d_in[1];
  const float* w1r = (const float*)d_in[2];
  const float* w1i = (const float*)d_in[3];
  const float* w2r = (const float*)d_in[4];
  const float* w2i = (const float*)d_in[5];
  const float* n1w = (const float*)d_in[6];
  const float* n1b = (const float*)d_in[7];
  const float* mw1 = (const float*)d_in[8];
  const float* mb1 = (const float*)d_in[9];
  const float* mw2 = (const float*)d_in[10];
  const float* mb2 = (const float*)d_in[11];
  const float* gw = (const float*)d_in[12];
  const float* gb = (const float*)d_in[13];
  const float* bw = (const float*)d_in[14];
  const float* bb = (const float*)d_in[15];
  float* out = (float*)d_out;

  // Workspace layout (floats); ~50.7 MB total.
  float* ws = (float*)d_ws;
  float* TW = ws;                    // 16384
  float* T2 = ws + 16384;            // 32768
  float* TIH = ws + 49152;           // 32768
  float* TIW = ws + 81920;           // 16384
  float* Xw = ws + 98304;            // 8,388,608 (reused as Yh)
  float* Xf = Xw + 8388608;          // 2,097,152
  float* Yc = Xf + 2097152;          // 2,097,152
  float* stats = Yc + 2097152;       // 32
  float* gbuf = stats + 32;          // 1024

  k_tables<<<1, 256, 0, stream>>>(ws);
  k_dftw<<<2048, 128, 0, stream>>>(x, TW, Xw);
  k_dfth<<<512, 128, 0, stream>>>(Xw, T2, Xf);
  k_spec<<<2048, 128, 0, stream>>>(Xf, w1r, w1i, w2r, w2i, Yc);
  k_idfth<<<2048, 128, 0, stream>>>(Yc, TIH, Xw /* Yh, reuse */);
  k_idftw<<<8192, 128, 0, stream>>>(Xw, TIW, out, stats);
  k_film<<<1, 512, 0, stream>>>(t, gw, gb, bw, bb, gbuf);
  k_mlp<<<8192, 128, 0, stream>>>(x, n1w, n1b, mw1, mb1, mw2, mb2, stats, out);
  k_final<<<32768, 256, 0, stream>>>(x, stats, gbuf, out);
}